// HCHA_2594160246969
// MI455X (gfx1250) — compile-verified
//
#include <hip/hip_runtime.h>
#include <hip/hip_bf16.h>
#include <math.h>

typedef __attribute__((ext_vector_type(2))) float v2f;
typedef __attribute__((ext_vector_type(8))) float v8f;
typedef const float __attribute__((address_space(1)))* gcf;
typedef float __attribute__((address_space(1)))* gf;

#define N_NODES 100000
#define N_EDGES 20000
#define N_NNZ   1000000
#define F_IN    128
#define F_HID   256
#define F_OUT   40

// ---------------------------------------------------------------------------
// Degree counting: D[n] = #incidences of node n, B[e] = #nodes in edge e
// ---------------------------------------------------------------------------
__global__ void hg_count(const int* __restrict__ ni, const int* __restrict__ ei,
                         float* __restrict__ D, float* __restrict__ B, int nnz) {
  int t = blockIdx.x * blockDim.x + threadIdx.x;
  if (t < nnz) {
    atomicAdd(&D[ni[t]], 1.0f);
    atomicAdd(&B[ei[t]], 1.0f);
  }
}

__global__ void hg_invert(float* __restrict__ D, float* __restrict__ B, int n, int e) {
  int t = blockIdx.x * blockDim.x + threadIdx.x;
  if (t < n) { float d = D[t]; D[t] = (d > 0.0f) ? (1.0f / d) : 0.0f; }
  if (t < e) { float v = B[t]; B[t] = (v > 0.0f) ? (1.0f / v) : 0.0f; }
}

// ---------------------------------------------------------------------------
// Vectorized gather + atomic scatter-add: dst[didx[i]] += src[sidx[i]]
// (F4 float4 chunks per row; atomics resolve in L2 — both scatter targets
//  [E,256]=20MB and [N,256]=100MB fit the 192MB L2)
// ---------------------------------------------------------------------------
__global__ void hg_scatter4(const float4* __restrict__ src,
                            const int* __restrict__ sidx, const int* __restrict__ didx,
                            float* __restrict__ dst, int nnz, int F4) {
  long long t = (long long)blockIdx.x * blockDim.x + threadIdx.x;
  if (t >= (long long)nnz * F4) return;
  int i = (int)(t / F4);
  int c = (int)(t - (long long)i * F4);
  float4 v = src[(size_t)sidx[i] * F4 + c];
  float* p = dst + ((size_t)didx[i] * F4 + c) * 4;
  atomicAdd(p + 0, v.x);
  atomicAdd(p + 1, v.y);
  atomicAdd(p + 2, v.z);
  atomicAdd(p + 3, v.w);
}

// ---------------------------------------------------------------------------
// Y[E x Fo] = (rowScale ⊙ A[E x Fi]) @ W[Fi x Fo]  via V_WMMA_F32_16X16X4_F32
// One wave computes a 16x64 strip (4 accumulator tiles, A reused 4x).
// Fi/Fo are compile-time: all B/A strides fold into global_load immediate
// offsets (no per-step address math); pointers bump once per 32-k chunk.
// Row scale (Binv) applied in the epilogue (D rows == A rows).
// grid = (E/16, ceil(Fo/256)), block = 128 (4 waves x 64 cols each)
// ---------------------------------------------------------------------------
template <int Fi, int Fo>
__global__ void __launch_bounds__(128)
hg_gemm_wmma(const float* __restrict__ A_, const float* __restrict__ rowScale_,
             const float* __restrict__ W_, float* __restrict__ Y_) {
  gcf A  = (gcf)(unsigned long long)A_;
  gcf W  = (gcf)(unsigned long long)W_;
  gcf RS = (gcf)(unsigned long long)rowScale_;
  gf  Y  = (gf)(unsigned long long)Y_;

  const int lane  = threadIdx.x & 31;
  const int wv    = threadIdx.x >> 5;
  const int nbase = (blockIdx.y * 4 + wv) * 64;     // first column of this wave
  if (nbase >= Fo) return;                          // wave-uniform early out
  const int l15     = lane & 15;
  const int kh      = lane >> 4;                    // 0: K pair {0,1}, 1: {2,3}
  const int rowTile = blockIdx.x * 16;              // E is a multiple of 16

  gcf ap = A + (size_t)(rowTile + l15) * Fi + 2 * kh;

  // Clamped (always in-bounds) per-tile B columns; garbage cols never stored.
  // For Fo % 64 == 0 instantiations the clamps are compile-time no-ops.
  const int col0 = nbase + 0 * 16 + l15;
  const int col1 = nbase + 1 * 16 + l15;
  const int col2 = nbase + 2 * 16 + l15;
  const int col3 = nbase + 3 * 16 + l15;
  gcf wb  = W + 2 * kh * Fo;
  gcf bp0 = wb + ((col0 < Fo) ? col0 : (Fo - 1));
  gcf bp1 = wb + ((col1 < Fo) ? col1 : (Fo - 1));
  gcf bp2 = wb + ((col2 < Fo) ? col2 : (Fo - 1));
  gcf bp3 = wb + ((col3 < Fo) ? col3 : (Fo - 1));

  constexpr int KSTEP = 32;                         // k-values per outer iter
  static_assert(Fi % KSTEP == 0, "Fi must be a multiple of KSTEP");

  v8f acc0 = {}, acc1 = {}, acc2 = {}, acc3 = {};
  for (int k0 = 0; k0 < Fi; k0 += KSTEP) {
#pragma unroll
    for (int k = 0; k < KSTEP; k += 4) {
      v2f a;
      a.x = ap[k];                                  // A[row][k0+k+2kh]
      a.y = ap[k + 1];
      v2f b0, b1, b2, b3;
      b0.x = bp0[k * Fo];  b0.y = bp0[(k + 1) * Fo];
      b1.x = bp1[k * Fo];  b1.y = bp1[(k + 1) * Fo];
      b2.x = bp2[k * Fo];  b2.y = bp2[(k + 1) * Fo];
      b3.x = bp3[k * Fo];  b3.y = bp3[(k + 1) * Fo];
      acc0 = __builtin_amdgcn_wmma_f32_16x16x4_f32(false, a, false, b0,
                                                   (short)0, acc0, false, false);
      acc1 = __builtin_amdgcn_wmma_f32_16x16x4_f32(false, a, false, b1,
                                                   (short)0, acc1, false, false);
      acc2 = __builtin_amdgcn_wmma_f32_16x16x4_f32(false, a, false, b2,
                                                   (short)0, acc2, false, false);
      acc3 = __builtin_amdgcn_wmma_f32_16x16x4_f32(false, a, false, b3,
                                                   (short)0, acc3, false, false);
    }
    ap  += KSTEP;
    bp0 += KSTEP * Fo;
    bp1 += KSTEP * Fo;
    bp2 += KSTEP * Fo;
    bp3 += KSTEP * Fo;
  }

  // Epilogue: row scale (Binv) — D layout: VGPR v -> row (v + 8*kh), col l15
  gcf rs = RS + rowTile + 8 * kh;
  float sc[8];
#pragma unroll
  for (int v = 0; v < 8; ++v) sc[v] = rs[v];        // contiguous -> b128 loads

  gf ybase = Y + (size_t)(rowTile + 8 * kh) * Fo;
#pragma unroll
  for (int v = 0; v < 8; ++v) {
    size_t roff = (size_t)v * Fo;
    if (col0 < Fo) ybase[roff + col0] = acc0[v] * sc[v];
    if (col1 < Fo) ybase[roff + col1] = acc1[v] * sc[v];
    if (col2 < Fo) ybase[roff + col2] = acc2[v] * sc[v];
    if (col3 < Fo) ybase[roff + col3] = acc3[v] * sc[v];
  }
}

// ---------------------------------------------------------------------------
// buf[n][f] = act(Dinv[n]*buf[n][f] + bias[f]) ; act = ELU (expm1) or identity
// ---------------------------------------------------------------------------
__global__ void hg_finalize(float* __restrict__ buf, const float* __restrict__ Dinv,
                            const float* __restrict__ bias, int n, int F, int act) {
  long long t = (long long)blockIdx.x * blockDim.x + threadIdx.x;
  if (t >= (long long)n * F) return;
  int r = (int)(t / F);
  int f = (int)(t - (long long)r * F);
  float v = Dinv[r] * buf[t] + bias[f];
  if (act) v = (v > 0.0f) ? v : expm1f(v);
  buf[t] = v;
}

static inline int cdiv_ll(long long a, int b) { return (int)((a + b - 1) / b); }

extern "C" void kernel_launch(void* const* d_in, const int* in_sizes, int n_in,
                              void* d_out, int out_size, void* d_ws, size_t ws_size,
                              hipStream_t stream) {
  (void)in_sizes; (void)n_in; (void)out_size;

  const float* x  = (const float*)d_in[0];   // [N, 128]
  const int*   ni = (const int*)d_in[1];     // [NNZ]
  const int*   ei = (const int*)d_in[2];     // [NNZ]
  const float* W1 = (const float*)d_in[3];   // [128, 256]
  const float* b1 = (const float*)d_in[4];   // [256]
  const float* W2 = (const float*)d_in[5];   // [256, 256]
  const float* b2 = (const float*)d_in[6];   // [256]
  const float* W3 = (const float*)d_in[7];   // [256, 40]
  const float* b3 = (const float*)d_in[8];   // [40]
  float* out = (float*)d_out;                // [N, 40]

  // workspace layout (all 16B-aligned given 256B-aligned d_ws)
  float* Dinv = (float*)d_ws;                                // N
  float* Binv = Dinv + N_NODES;                              // E
  float* M0   = Binv + N_EDGES;                              // E * 256
  float* Yb   = M0 + (size_t)N_EDGES * F_HID;                // E * 256
  float* H    = Yb + (size_t)N_EDGES * F_HID;                // N * 256
  size_t need = ((size_t)N_NODES + N_EDGES +
                 2 * (size_t)N_EDGES * F_HID +
                 (size_t)N_NODES * F_HID) * sizeof(float);
  if (ws_size < need) return;  // workspace too small; cannot proceed

  // --- degree norms: Dinv = 1/D (0 if empty), Binv = 1/B ---
  hipMemsetAsync(Dinv, 0, sizeof(float) * (N_NODES + N_EDGES), stream);
  hg_count<<<cdiv_ll(N_NNZ, 256), 256, 0, stream>>>(ni, ei, Dinv, Binv, N_NNZ);
  hg_invert<<<cdiv_ll(N_NODES, 256), 256, 0, stream>>>(Dinv, Binv, N_NODES, N_EDGES);

  // ======================= layer 1: 128 -> 256, ELU =======================
  hipMemsetAsync(M0, 0, sizeof(float) * (size_t)N_EDGES * F_IN, stream);
  hg_scatter4<<<cdiv_ll((long long)N_NNZ * (F_IN / 4), 256), 256, 0, stream>>>(
      (const float4*)x, ni, ei, M0, N_NNZ, F_IN / 4);
  {
    dim3 g(N_EDGES / 16, (F_HID + 255) / 256);
    hg_gemm_wmma<F_IN, F_HID><<<g, 128, 0, stream>>>(M0, Binv, W1, Yb);
  }
  hipMemsetAsync(H, 0, sizeof(float) * (size_t)N_NODES * F_HID, stream);
  hg_scatter4<<<cdiv_ll((long long)N_NNZ * (F_HID / 4), 256), 256, 0, stream>>>(
      (const float4*)Yb, ei, ni, H, N_NNZ, F_HID / 4);
  hg_finalize<<<cdiv_ll((long long)N_NODES * F_HID, 256), 256, 0, stream>>>(
      H, Dinv, b1, N_NODES, F_HID, 1);

  // ======================= layer 2: 256 -> 256, ELU =======================
  hipMemsetAsync(M0, 0, sizeof(float) * (size_t)N_EDGES * F_HID, stream);
  hg_scatter4<<<cdiv_ll((long long)N_NNZ * (F_HID / 4), 256), 256, 0, stream>>>(
      (const float4*)H, ni, ei, M0, N_NNZ, F_HID / 4);
  {
    dim3 g(N_EDGES / 16, (F_HID + 255) / 256);
    hg_gemm_wmma<F_HID, F_HID><<<g, 128, 0, stream>>>(M0, Binv, W2, Yb);
  }
  hipMemsetAsync(H, 0, sizeof(float) * (size_t)N_NODES * F_HID, stream);
  hg_scatter4<<<cdiv_ll((long long)N_NNZ * (F_HID / 4), 256), 256, 0, stream>>>(
      (const float4*)Yb, ei, ni, H, N_NNZ, F_HID / 4);
  hg_finalize<<<cdiv_ll((long long)N_NODES * F_HID, 256), 256, 0, stream>>>(
      H, Dinv, b2, N_NODES, F_HID, 1);

  // ======================= layer 3: 256 -> 40, linear =====================
  hipMemsetAsync(M0, 0, sizeof(float) * (size_t)N_EDGES * F_HID, stream);
  hg_scatter4<<<cdiv_ll((long long)N_NNZ * (F_HID / 4), 256), 256, 0, stream>>>(
      (const float4*)H, ni, ei, M0, N_NNZ, F_HID / 4);
  {
    dim3 g(N_EDGES / 16, (F_OUT + 255) / 256);
    hg_gemm_wmma<F_HID, F_OUT><<<g, 128, 0, stream>>>(M0, Binv, W3, Yb);
  }
  hipMemsetAsync(out, 0, sizeof(float) * (size_t)N_NODES * F_OUT, stream);
  hg_scatter4<<<cdiv_ll((long long)N_NNZ * (F_OUT / 4), 256), 256, 0, stream>>>(
      (const float4*)Yb, ei, ni, out, N_NNZ, F_OUT / 4);
  hg_finalize<<<cdiv_ll((long long)N_NODES * F_OUT, 256), 256, 0, stream>>>(
      out, Dinv, b3, N_NODES, F_OUT, 0);
}